// subnet_GCN_17411797418191
// MI455X (gfx1250) — compile-verified
//
#include <hip/hip_runtime.h>
#include <hip/hip_bf16.h>

typedef __attribute__((ext_vector_type(2))) float v2f;
typedef __attribute__((ext_vector_type(8))) float v8f;

__device__ __forceinline__ float leaky(float v) { return v > 0.f ? v : 0.1f * v; }

// ---- 1. init: deg = 1.0 (self loop), acc[0..15] = 0 ----
__global__ void k_init(float* deg, float* acc, int N) {
    int i = blockIdx.x * blockDim.x + threadIdx.x;
    if (i < N) deg[i] = 1.0f;
    if (i < 16) acc[i] = 0.0f;
}

// ---- 2. degree accumulation over edges ----
__global__ void k_deg(const int* dst, float* deg, int E) {
    int e = blockIdx.x * blockDim.x + threadIdx.x;
    if (e < E) atomicAdd(&deg[dst[e]], 1.0f);
}

// ---- 3. dinv = rsqrt(deg) (in place), s1 self-loop init = dinv^2 * x ----
__global__ void k_dinv_s1(float* deg_dinv, const float* x, float* s1, int N) {
    int n = blockIdx.x * blockDim.x + threadIdx.x;
    if (n < N) {
        float r = rsqrtf(deg_dinv[n]);   // deg >= 1 always (self loop)
        deg_dinv[n] = r;
        s1[n] = r * r * x[n];
    }
}

// ---- 4. layer-1 scalar aggregation over edges: s1[dst] += norm * x[src] ----
__global__ void k_s1_edges(const int* src, const int* dst, const float* dinv,
                           const float* x, float* s1, int E) {
    int e = blockIdx.x * blockDim.x + threadIdx.x;
    if (e < E) {
        int s = src[e], d = dst[e];
        atomicAdd(&s1[d], dinv[s] * dinv[d] * x[s]);
    }
}

// ---- 5. fused layer-1 feature expansion + layer-2 GEMM via WMMA f32 16x16x4 ----
// h1[m,k] = leaky(s1[m]*W1[k] + b1[k]) generated on the fly as the A operand.
// y2 = h1 @ W2  [N,32]; agg2 initialized with self-loop term dinv^2 * y2.
__global__ void k_gemm2(const float* s1, const float* dinv, const float* W1,
                        const float* b1, const float* W2, float* y2, float* agg2,
                        int N) {
    int w    = blockIdx.x * (blockDim.x >> 5) + (threadIdx.x >> 5); // wave id
    int mt   = w >> 1;          // 16-row tile
    int nt   = w & 1;           // 16-col tile (32 cols total)
    if (mt * 16 >= N) return;   // wave-uniform
    int lane = threadIdx.x & 31;
    int l16  = lane & 15;
    int half = lane >> 4;
    int m0 = mt * 16, n0 = nt * 16;

    float sv = s1[m0 + l16];    // this lane's A-row scalar
    v8f c = {};
    #pragma unroll
    for (int k = 0; k < 64; k += 4) {
        v2f a, b;
        #pragma unroll
        for (int j = 0; j < 2; ++j) {
            int kk = k + half * 2 + j;
            float h = sv * W1[kk] + b1[kk];
            a[j] = leaky(h);                       // A: M=l16, K=kk
            b[j] = W2[kk * 32 + n0 + l16];         // B: K=kk, N=n0+l16
        }
        c = __builtin_amdgcn_wmma_f32_16x16x4_f32(false, a, false, b,
                                                  (short)0, c, false, false);
    }
    #pragma unroll
    for (int r = 0; r < 8; ++r) {
        int row = m0 + r + half * 8;               // C: VGPR r -> M = r (+8)
        int col = n0 + l16;
        float v  = c[r];
        float dv = dinv[row];
        y2[row * 32 + col]   = v;
        agg2[row * 32 + col] = dv * dv * v;        // self-loop contribution
    }
}

// ---- 6. layer-2 aggregation: lane j of each wave handles feature j of edge e ----
__global__ void k_agg2_edges(const int* src, const int* dst, const float* dinv,
                             const float* y2, float* agg2, int E) {
    long long t = (long long)blockIdx.x * blockDim.x + threadIdx.x;
    if (t >= (long long)E * 32) return;
    int e = (int)(t >> 5);
    int j = (int)(t & 31);
    int s = src[e], d = dst[e];                    // uniform across wave
    float nm = dinv[s] * dinv[d];
    atomicAdd(&agg2[d * 32 + j], nm * y2[s * 32 + j]);
}

// ---- 7. h2 = leaky(agg2 + b2), in place ----
__global__ void k_h2(float* agg2, const float* b2, int N) {
    int i = blockIdx.x * blockDim.x + threadIdx.x;
    if (i < N * 32) {
        float v = agg2[i] + b2[i & 31];
        agg2[i] = leaky(v);
    }
}

// ---- 8. layer-3 GEMM via WMMA: y3 = h2 @ W3 (cols padded 10 -> 16) ----
__global__ void k_gemm3(const float* h2, const float* W3, float* y3, int N) {
    int w = blockIdx.x * (blockDim.x >> 5) + (threadIdx.x >> 5);
    if (w * 16 >= N) return;    // wave-uniform
    int lane = threadIdx.x & 31;
    int l16  = lane & 15;
    int half = lane >> 4;
    int m0 = w * 16;

    v8f c = {};
    #pragma unroll
    for (int k = 0; k < 32; k += 4) {
        v2f a, b;
        #pragma unroll
        for (int j = 0; j < 2; ++j) {
            int kk = k + half * 2 + j;
            a[j] = h2[(m0 + l16) * 32 + kk];                 // A: M=l16, K=kk
            b[j] = (l16 < 10) ? W3[kk * 10 + l16] : 0.0f;    // B padded to 16
        }
        c = __builtin_amdgcn_wmma_f32_16x16x4_f32(false, a, false, b,
                                                  (short)0, c, false, false);
    }
    #pragma unroll
    for (int r = 0; r < 8; ++r) {
        int row = m0 + r + half * 8;
        y3[row * 16 + l16] = c[r];
    }
}

// ---- 9. final: out = (1/N) * sum over (edges + self loops) of norm * y3[src] ----
// item index i < E: real edge; i >= E: self loop for node i-E.
__global__ void k_final_reduce(const int* src, const int* dst, const float* dinv,
                               const float* y3, float* acc, int E, int N) {
    int j    = threadIdx.x & 15;     // feature
    int slot = threadIdx.x >> 4;     // 0..15 item slots per block (blockDim=256)
    int total = E + N;
    float local = 0.0f;
    int stride = gridDim.x * 16;
    for (int i = blockIdx.x * 16 + slot; i < total; i += stride) {
        int s, d;
        if (i < E) { s = src[i]; d = dst[i]; }
        else       { s = i - E;  d = s; }
        local += dinv[s] * dinv[d] * y3[s * 16 + j];
    }
    __shared__ float red[256];
    red[threadIdx.x] = local;
    __syncthreads();
    for (int off = 128; off >= 16; off >>= 1) {
        if (threadIdx.x < off) red[threadIdx.x] += red[threadIdx.x + off];
        __syncthreads();
    }
    if (threadIdx.x < 16) atomicAdd(&acc[threadIdx.x], red[threadIdx.x]);
}

// ---- 10. write output: mean + b3 ----
__global__ void k_out(const float* acc, const float* b3, float* out, int N) {
    int j = threadIdx.x;
    if (j < 10) out[j] = acc[j] * (1.0f / (float)N) + b3[j];
}

extern "C" void kernel_launch(void* const* d_in, const int* in_sizes, int n_in,
                              void* d_out, int out_size, void* d_ws, size_t ws_size,
                              hipStream_t stream) {
    const float* x  = (const float*)d_in[0];
    const int*   ei = (const int*)d_in[1];
    const float* W1 = (const float*)d_in[2];
    const float* b1 = (const float*)d_in[3];
    const float* W2 = (const float*)d_in[4];
    const float* b2 = (const float*)d_in[5];
    const float* W3 = (const float*)d_in[6];
    const float* b3 = (const float*)d_in[7];

    const int N = in_sizes[0];
    const int E = in_sizes[1] / 2;
    const int* src = ei;
    const int* dst = ei + E;

    float* ws   = (float*)d_ws;
    float* dinv = ws;                       // N floats (deg, then dinv in place)
    float* s1   = ws + (size_t)N;           // N
    float* y2   = ws + 2  * (size_t)N;      // 32N
    float* agg2 = ws + 34 * (size_t)N;      // 32N (becomes h2 in place)
    float* y3   = ws + 66 * (size_t)N;      // 16N
    float* acc  = ws + 82 * (size_t)N;      // 16

    const int B = 256;
    int nblkN  = (N + B - 1) / B;
    int nblkE  = (E + B - 1) / B;

    k_init<<<nblkN, B, 0, stream>>>(dinv, acc, N);
    k_deg<<<nblkE, B, 0, stream>>>(dst, dinv, E);
    k_dinv_s1<<<nblkN, B, 0, stream>>>(dinv, x, s1, N);
    k_s1_edges<<<nblkE, B, 0, stream>>>(src, dst, dinv, x, s1, E);

    // layer 2 GEMM: ceil(N/16) row tiles * 2 col tiles, 4 waves (tiles) / block
    int waves2 = ((N + 15) / 16) * 2;
    k_gemm2<<<(waves2 + 3) / 4, 128, 0, stream>>>(s1, dinv, W1, b1, W2, y2, agg2, N);

    long long tE32 = (long long)E * 32;
    k_agg2_edges<<<(int)((tE32 + B - 1) / B), B, 0, stream>>>(src, dst, dinv, y2, agg2, E);

    k_h2<<<(N * 32 + B - 1) / B, B, 0, stream>>>(agg2, b2, N);

    int waves3 = (N + 15) / 16;
    k_gemm3<<<(waves3 + 3) / 4, 128, 0, stream>>>(agg2, W3, y3, N);

    k_final_reduce<<<2048, 256, 0, stream>>>(src, dst, dinv, y3, acc, E, N);
    k_out<<<1, 16, 0, stream>>>(acc, b3, (float*)d_out, N);
}